// EntityAwareAttention_12197707120951
// MI455X (gfx1250) — compile-verified
//
#include <hip/hip_runtime.h>

typedef __attribute__((ext_vector_type(2))) float f32x2;
typedef __attribute__((ext_vector_type(8))) float f32x8;

#define BDIM   2048
#define LDIM   128
#define H2DIM  600
#define PDIM   50
#define ADIM   50
#define TDIM   3
#define K1DIM  700            // H2 + 2P, divisible by 4
#define K2DIM  2400           // 4*H2
#define WSTRIDE 702           // padded LDS row stride (dwords) to spread banks

// LDS layout (floats)
#define SW_OFF   0
#define CAT_OFF  (64 * WSTRIDE)
#define U2_OFF   (CAT_OFF + K2DIM)
#define VV_OFF   (U2_OFF + 64)
#define SC_OFF   (VV_OFF + 64)
#define RED_OFF  (SC_OFF + LDIM)
#define SMEM_FLOATS (RED_OFF + 64)

// One 16x4 A-fragment against 4 B-tiles (n0 = 0,16,32,48)
#define GEMM_STEP(AFRAG, KLO)                                                                              \
    {                                                                                                      \
        f32x2 bf0 = *(const f32x2*)(&sw[(0  + ln) * WSTRIDE + (KLO)]);                                     \
        f32x2 bf1 = *(const f32x2*)(&sw[(16 + ln) * WSTRIDE + (KLO)]);                                     \
        f32x2 bf2 = *(const f32x2*)(&sw[(32 + ln) * WSTRIDE + (KLO)]);                                     \
        f32x2 bf3 = *(const f32x2*)(&sw[(48 + ln) * WSTRIDE + (KLO)]);                                     \
        acc0 = __builtin_amdgcn_wmma_f32_16x16x4_f32(false, (AFRAG), false, bf0, (short)0, acc0, false, false); \
        acc1 = __builtin_amdgcn_wmma_f32_16x16x4_f32(false, (AFRAG), false, bf1, (short)0, acc1, false, false); \
        acc2 = __builtin_amdgcn_wmma_f32_16x16x4_f32(false, (AFRAG), false, bf2, (short)0, acc2, false, false); \
        acc3 = __builtin_amdgcn_wmma_f32_16x16x4_f32(false, (AFRAG), false, bf3, (short)0, acc3, false, false); \
    }

__global__ __launch_bounds__(256)
void ent_attn_kernel(const float* __restrict__ hidden,
                     const float* __restrict__ pos1,
                     const float* __restrict__ pos2,
                     const int*   __restrict__ e1idx,
                     const int*   __restrict__ e2idx,
                     const float* __restrict__ W_hid,
                     const float* __restrict__ W_ent,
                     const float* __restrict__ lt,
                     const float* __restrict__ vvec,
                     float*       __restrict__ out)
{
    extern __shared__ float smem[];
    float* sw   = smem + SW_OFF;    // [64][WSTRIDE] padded W_hid
    float* scat = smem + CAT_OFF;   // [2400] entity concat vector
    float* su2  = smem + U2_OFF;    // [64]   u2, zero-padded
    float* svv  = smem + VV_OFF;    // [64]   v,  zero-padded
    float* ssc  = smem + SC_OFF;    // [128]  scores -> alpha
    float* sred = smem + RED_OFF;   // [64]   reduction scratch

    const int b    = blockIdx.x;
    const int tid  = threadIdx.x;
    const int lane = tid & 31;
    const int wv   = tid >> 5;      // wave id 0..7 == M tile
    const int ln   = lane & 15;
    const int half = lane >> 4;

    const int e1 = e1idx[b];
    const int e2 = e2idx[b];
    const float* e1row = hidden + ((size_t)b * LDIM + e1) * H2DIM;
    const float* e2row = hidden + ((size_t)b * LDIM + e2) * H2DIM;

    // ---- Phase 0: stage padded W_hid into LDS; v + u2 pad; latent-type dots ----
    for (int idx = tid; idx < 64 * K1DIM; idx += 256) {
        int r = idx / K1DIM;
        int c = idx - r * K1DIM;
        sw[r * WSTRIDE + c] = (r < ADIM) ? W_hid[r * K1DIM + c] : 0.0f;
    }
    if (tid < 64) {
        svv[tid] = (tid < ADIM) ? vvec[tid] : 0.0f;
        su2[tid] = 0.0f;
    }

    float p[6] = {0.f, 0.f, 0.f, 0.f, 0.f, 0.f};
    for (int d = tid; d < H2DIM; d += 256) {
        float h1 = e1row[d], h2 = e2row[d];
        float l0 = lt[0 * H2DIM + d];
        float l1 = lt[1 * H2DIM + d];
        float l2 = lt[2 * H2DIM + d];
        p[0] += h1 * l0; p[1] += h1 * l1; p[2] += h1 * l2;
        p[3] += h2 * l0; p[4] += h2 * l1; p[5] += h2 * l2;
    }
#pragma unroll
    for (int m = 1; m < 32; m <<= 1) {
#pragma unroll
        for (int j = 0; j < 6; ++j) p[j] += __shfl_xor(p[j], m, 32);
    }
    if (lane == 0) {
#pragma unroll
        for (int j = 0; j < 6; ++j) sred[j * 8 + wv] = p[j];
    }
    __syncthreads();
    if (tid < 6) {
        float s = 0.f;
#pragma unroll
        for (int w = 0; w < 8; ++w) s += sred[tid * 8 + w];
        sred[48 + tid] = s;
    }
    __syncthreads();

    // ---- Phase 1: latent-type softmax weights (all threads, tiny) + cat build ----
    {
        float s10 = sred[48 + 0], s11 = sred[48 + 1], s12 = sred[48 + 2];
        float s20 = sred[48 + 3], s21 = sred[48 + 4], s22 = sred[48 + 5];
        float m1 = fmaxf(fmaxf(s10, s11), s12);
        float m2 = fmaxf(fmaxf(s20, s21), s22);
        float a10 = expf(s10 - m1), a11 = expf(s11 - m1), a12 = expf(s12 - m1);
        float a20 = expf(s20 - m2), a21 = expf(s21 - m2), a22 = expf(s22 - m2);
        float i1 = 1.0f / (a10 + a11 + a12);
        float i2 = 1.0f / (a20 + a21 + a22);
        float w10 = a10 * i1, w11 = a11 * i1, w12 = a12 * i1;
        float w20 = a20 * i2, w21 = a21 * i2, w22 = a22 * i2;
        for (int d = tid; d < H2DIM; d += 256) {
            float l0 = lt[0 * H2DIM + d];
            float l1 = lt[1 * H2DIM + d];
            float l2 = lt[2 * H2DIM + d];
            scat[d]             = e1row[d];
            scat[H2DIM + d]     = w10 * l0 + w11 * l1 + w12 * l2;
            scat[2 * H2DIM + d] = e2row[d];
            scat[3 * H2DIM + d] = w20 * l0 + w21 * l1 + w22 * l2;
        }
    }
    __syncthreads();

    // ---- Phase 2: u2[a] = cat . W_ent[a,:]  (4 lanes per output a) ----
    if (tid < 4 * ADIM) {
        int a  = tid >> 2;
        int kl = tid & 3;
        const float* wrow = W_ent + (size_t)a * K2DIM;
        float s = 0.f;
        for (int k = kl; k < K2DIM; k += 4) s += scat[k] * wrow[k];
        s += __shfl_xor(s, 1, 32);
        s += __shfl_xor(s, 2, 32);
        if (kl == 0) su2[a] = s;
    }
    __syncthreads();

    // ---- Phase 3: GEMM u1 = X @ W_hid^T via V_WMMA_F32_16X16X4_F32 ----
    // wave wv handles M rows [wv*16, wv*16+16), 4 N tiles (n0 = 0,16,32,48).
    // K loop split into uniform segments so the concat needs no per-lane branches:
    //   [0,600) hidden | [600,648) pos1 | kb==648 straddle | [652,700) pos2
    const int mrow = wv * 16 + ln;
    const float* hrow  = hidden + ((size_t)b * LDIM + mrow) * H2DIM;
    const float* p1row = pos1   + ((size_t)b * LDIM + mrow) * PDIM;
    const float* p2row = pos2   + ((size_t)b * LDIM + mrow) * PDIM;

    f32x8 acc0 = {}, acc1 = {}, acc2 = {}, acc3 = {};

    for (int kb = 0; kb < H2DIM; kb += 4) {
        const int klo = kb + 2 * half;
        // WGP-scope prefetch of the sequential A stream (locality 3 -> near caches)
        if ((kb & 63) == 0) __builtin_prefetch(hrow + klo + 256, 0, 3);
        f32x2 afrag = *(const f32x2*)(hrow + klo);
        GEMM_STEP(afrag, klo);
    }
    for (int kb = H2DIM; kb < 648; kb += 4) {
        const int klo = kb + 2 * half;
        f32x2 afrag = *(const f32x2*)(p1row + (klo - H2DIM));
        GEMM_STEP(afrag, klo);
    }
    {   // straddling step kb == 648: half 0 -> pos1[48..49], half 1 -> pos2[0..1]
        const int klo = 648 + 2 * half;
        const float* base = half ? (p2row - (H2DIM + PDIM)) : (p1row - H2DIM);
        f32x2 afrag = *(const f32x2*)(base + klo);
        GEMM_STEP(afrag, klo);
    }
    for (int kb = 652; kb < K1DIM; kb += 4) {
        const int klo = kb + 2 * half;
        f32x2 afrag = *(const f32x2*)(p2row + (klo - (H2DIM + PDIM)));
        GEMM_STEP(afrag, klo);
    }

    // ---- Phase 4: scores[m] = sum_n tanh(u1 + u2[n]) * v[n] ----
    {
        const float u2_0 = su2[0  + ln], v_0 = svv[0  + ln];
        const float u2_1 = su2[16 + ln], v_1 = svv[16 + ln];
        const float u2_2 = su2[32 + ln], v_2 = svv[32 + ln];
        const float u2_3 = su2[48 + ln], v_3 = svv[48 + ln];
#pragma unroll
        for (int r = 0; r < 8; ++r) {
            float x = tanhf(acc0[r] + u2_0) * v_0
                    + tanhf(acc1[r] + u2_1) * v_1
                    + tanhf(acc2[r] + u2_2) * v_2
                    + tanhf(acc3[r] + u2_3) * v_3;
            x += __shfl_xor(x, 1, 32);
            x += __shfl_xor(x, 2, 32);
            x += __shfl_xor(x, 4, 32);
            x += __shfl_xor(x, 8, 32);
            if (ln == 0) ssc[wv * 16 + r + 8 * half] = x;  // D layout: m = r + 8*(lane/16)
        }
    }
    __syncthreads();

    // ---- Phase 5: softmax over L=128 ----
    if (tid == 0) {
        float mx = -INFINITY;
        for (int l = 0; l < LDIM; ++l) mx = fmaxf(mx, ssc[l]);
        sred[56] = mx;
    }
    __syncthreads();
    if (tid < LDIM) ssc[tid] = expf(ssc[tid] - sred[56]);
    __syncthreads();
    if (tid == 0) {
        float s = 0.f;
        for (int l = 0; l < LDIM; ++l) s += ssc[l];
        sred[57] = 1.0f / s;
    }
    __syncthreads();
    if (tid < LDIM) ssc[tid] *= sred[57];
    __syncthreads();

    // ---- Phase 6: z[d] = sum_l alpha[l] * hidden[b,l,d] (coalesced over d) ----
    const float* hb = hidden + (size_t)b * LDIM * H2DIM;
    for (int d = tid; d < H2DIM; d += 256) {
        float acc = 0.f;
        for (int l = 0; l < LDIM; ++l) acc = fmaf(ssc[l], hb[(size_t)l * H2DIM + d], acc);
        out[(size_t)b * H2DIM + d] = acc;
    }
}

extern "C" void kernel_launch(void* const* d_in, const int* in_sizes, int n_in,
                              void* d_out, int out_size, void* d_ws, size_t ws_size,
                              hipStream_t stream) {
    (void)in_sizes; (void)n_in; (void)d_ws; (void)ws_size; (void)out_size;
    const float* hidden = (const float*)d_in[0];
    const float* pos1   = (const float*)d_in[1];
    const float* pos2   = (const float*)d_in[2];
    const int*   e1     = (const int*)d_in[3];
    const int*   e2     = (const int*)d_in[4];
    const float* W_hid  = (const float*)d_in[5];
    const float* W_ent  = (const float*)d_in[6];
    const float* lt     = (const float*)d_in[7];
    const float* vv     = (const float*)d_in[8];
    float* out = (float*)d_out;

    const size_t smem = (size_t)SMEM_FLOATS * sizeof(float);
    hipFuncSetAttribute((const void*)ent_attn_kernel,
                        hipFuncAttributeMaxDynamicSharedMemorySize, (int)smem);
    ent_attn_kernel<<<BDIM, 256, smem, stream>>>(hidden, pos1, pos2, e1, e2,
                                                 W_hid, W_ent, lt, vv, out);
}